// PCEN_2783138807955
// MI455X (gfx1250) — compile-verified
//
#include <hip/hip_runtime.h>
#include <math.h>

// PCEN on MI455X (gfx1250): EMA scan expressed as 16x16 lower-triangular
// Toeplitz matmul per time tile, executed with chained V_WMMA_F32_16X16X4_F32.
// Wave32: each wave owns 16 sequence rows; D-matrix N dim = time-in-tile so
// stores coalesce (2x64B per store). T split into 8 chunks with 768-step
// warmup (decay 0.975^768 ~ 4e-9 -> exact in f32) giving 4096 waves; warmup
// re-reads hit the 192MB L2 (whole 128MiB input is L2-resident; output stores
// are non-temporal so they don't evict it).

typedef __attribute__((ext_vector_type(2))) float v2f;
typedef __attribute__((ext_vector_type(8))) float v8f;

#define PCEN_S     0.025f
#define PCEN_W     0.975f   // 1 - S
#define PCEN_ALPHA 0.98f
#define PCEN_EPS   1e-6f
#define TDIM   4096
#define CHUNKS 8
#define CLEN   (TDIM / CHUNKS)   // 512
#define WARM   768

__global__ __launch_bounds__(128) void pcen_wmma_scan_kernel(
    const float* __restrict__ x, float* __restrict__ out, int R) {
  const int lane = threadIdx.x & 31;
  const int wid  = blockIdx.x * (blockDim.x >> 5) + (threadIdx.x >> 5);
  const int wavesPerChunk = R >> 4;
  const int g     = wid % wavesPerChunk;
  const int chunk = wid / wavesPerChunk;
  const int R0    = g << 4;

  const int i    = lane & 15;   // time index within tile (D's N) == A's M row
  const int half = lane >> 4;

  // carry decay within a tile: smooth[i] gets w^(i+1) * carry
  const float wpow = __powf(PCEN_W, (float)(i + 1));

  // Constant B = M^T per K-block. M[i][j] = (j<=i) ? S*w^(i-j) : 0.
  // 32-bit 4x16 B layout: VGPR r, lane L -> K = 2*(L/16)+r, N = L%16.
  v2f Bk[4];
  #pragma unroll
  for (int kb = 0; kb < 4; ++kb) {
    #pragma unroll
    for (int r = 0; r < 2; ++r) {
      const int j = 4 * kb + 2 * half + r;
      Bk[kb][r] = (j <= i) ? PCEN_S * __powf(PCEN_W, (float)(i - j)) : 0.0f;
    }
  }

  const int tmain = chunk * CLEN;
  const int tend  = tmain + CLEN;
  int tw = tmain - WARM; if (tw < 0) tw = 0;   // tw==0 => exact init

  // A-layout loads come from row (lane&15); D-layout rows are r + 8*half.
  const float* rowA = x + (size_t)(R0 + i) * TDIM;
  const size_t rowD = (size_t)(R0 + 8 * half) * TDIM;

  // carry[r] for sequence n = r + 8*half. Init so smooth[tw] == x[tw].
  float carry[8];
  #pragma unroll
  for (int r = 0; r < 8; ++r)
    carry[r] = x[rowD + (size_t)r * TDIM + tw];

  const int   srcLane = (lane & 16) + 15;        // lane holding (n, i=15)
  const float SQD     = 1.41421356237309515f;    // sqrt(DELTA=2)

  for (int t0 = tw; t0 < tend; t0 += 16) {
    // prefetch ~8 tiles ahead on the streaming A reads (cold HBM pass)
    __builtin_prefetch(rowA + t0 + 128, 0, 1);

    // ---- tile scan: D[n][i] = sum_j M[i][j] * x[n][t0+j], 4 chained WMMAs
    v8f acc = {};
    #pragma unroll
    for (int kb = 0; kb < 4; ++kb) {
      v2f a = *(const v2f*)(rowA + t0 + 4 * kb + 2 * half);
      acc = __builtin_amdgcn_wmma_f32_16x16x4_f32(
          false, a, false, Bk[kb], (short)0, acc, false, false);
    }

    // add carry term; new carry = smooth at i==15 (lanes 15 / 31)
    float sm[8];
    #pragma unroll
    for (int r = 0; r < 8; ++r) sm[r] = acc[r] + wpow * carry[r];
    #pragma unroll
    for (int r = 0; r < 8; ++r) carry[r] = __shfl(sm[r], srcLane, 32);

    if (t0 >= tmain) {   // wave-uniform: outputs only in the main region
      #pragma unroll
      for (int r = 0; r < 8; ++r) {
        const size_t idx = rowD + (size_t)r * TDIM + (size_t)(t0 + i);
        const float xv  = x[idx];                          // L0/L2 hit
        const float v   = sm[r] + PCEN_EPS;                // >= 1e-6: no denorm guard needed
        // v^-alpha via raw hardware log2/exp2 (single-instruction trans ops)
        const float pin = __builtin_amdgcn_exp2f(-PCEN_ALPHA * __builtin_amdgcn_logf(v));
        const float y   = __builtin_amdgcn_sqrtf(fmaf(xv, pin, 2.0f)) - SQD;
        __builtin_nontemporal_store(y, &out[idx]);         // don't evict input from L2
      }
    }
  }
}

extern "C" void kernel_launch(void* const* d_in, const int* in_sizes, int n_in,
                              void* d_out, int out_size, void* d_ws, size_t ws_size,
                              hipStream_t stream) {
  const float* x = (const float*)d_in[0];
  float* out = (float*)d_out;
  const int total = in_sizes[0];          // B*C*T = 64*128*4096
  const int R = total / TDIM;             // 8192 independent sequences
  const int waves = (R / 16) * CHUNKS;    // 4096
  const int threads = 128;                // 4 waves per block
  const int blocks = waves / (threads / 32);
  pcen_wmma_scan_kernel<<<blocks, threads, 0, stream>>>(x, out, R);
}